// Memory_Pooling_Layer_71966472011820
// MI455X (gfx1250) — compile-verified
//
#include <hip/hip_runtime.h>

// ---------------------------------------------------------------------------
// Memory_Pooling_Layer, MI455X (gfx1250, wave32).
//
// Key identity: reference applies softmax over a SINGLETON axis -> C == 1
// exactly. So:
//   new_node_set[b,k,:] = LeakyReLU( (sum_n node[b,n,:]) @ W + bias )   (k-invariant)
//   new_adj[b,k,j]      = sum_{n,m} adj[b,n,m]                          (k,j-invariant)
// centroids / agg_w / agg_b are dead. Problem is HBM-bound (256MB adj read).
// ---------------------------------------------------------------------------

typedef float v2f __attribute__((ext_vector_type(2)));
typedef float v8f __attribute__((ext_vector_type(8)));

#define BB   16
#define NN   2048
#define DD   64
#define NOUT 128

// Workspace layout (floats):
//   [0, 16)            adj_sum[b]
//   [16, 16+4096)      adj block partials [b][256]
//   [4112, 4112+1024)  S[b][d]  (column sums of node_set)
//   [5136, 5136+1024)  Y[b][d]  (LeakyReLU(S@W + bias))
#define WS_ADJSUM  0
#define WS_ADJPART 16
#define WS_S       4112
#define WS_Y       5136

// --- Kernel 1: per-block partial sums of adj (coalesced float4) -------------
__global__ void adj_partial_kernel(const float* __restrict__ adj,
                                   float* __restrict__ part) {
    const int b   = blockIdx.y;           // batch
    const int blk = blockIdx.x;           // 256 blocks per batch
    const int t   = threadIdx.x;          // 256 threads
    // per batch: 2048*2048 = 4,194,304 floats = 1,048,576 float4
    // per block: 4096 float4 ; per thread: 16 float4 (stride-256, coalesced)
    const float4* a4 = reinterpret_cast<const float4*>(adj) +
                       (size_t)b * 1048576u + (size_t)blk * 4096u;
    float acc = 0.0f;
#pragma unroll
    for (int i = 0; i < 16; ++i) {
        float4 v = a4[t + i * 256];
        acc += (v.x + v.y) + (v.z + v.w);
    }
    __shared__ float red[256];
    red[t] = acc;
    __syncthreads();
    for (int s = 128; s > 0; s >>= 1) {
        if (t < s) red[t] += red[t + s];
        __syncthreads();
    }
    if (t == 0) part[b * 256 + blk] = red[0];
}

// --- Kernel 2: reduce 256 partials per batch (deterministic tree) -----------
__global__ void adj_final_kernel(const float* __restrict__ part,
                                 float* __restrict__ adj_sum) {
    const int b = blockIdx.x;
    const int t = threadIdx.x;            // 256
    __shared__ float red[256];
    red[t] = part[b * 256 + t];
    __syncthreads();
    for (int s = 128; s > 0; s >>= 1) {
        if (t < s) red[t] += red[t + s];
        __syncthreads();
    }
    if (t == 0) adj_sum[b] = red[0];
}

// --- Kernel 3: column sums of node_set: S[b,d] = sum_n node[b,n,d] ----------
__global__ void node_rowsum_kernel(const float* __restrict__ node,
                                   float* __restrict__ S) {
    const int b  = blockIdx.x;            // 16 blocks
    const int t  = threadIdx.x;           // 1024 threads = 16 row-groups x 64 cols
    const int d  = t & 63;
    const int rg = t >> 6;                // 0..15
    const float* base = node + (size_t)b * (NN * DD);
    float acc = 0.0f;
    for (int r = rg; r < NN; r += 16)     // 128 rows per thread, coalesced 64-wide
        acc += base[r * DD + d];
    __shared__ float red[1024];
    red[t] = acc;
    __syncthreads();
    if (rg == 0) {
        float s = 0.0f;
#pragma unroll
        for (int g = 0; g < 16; ++g) s += red[g * 64 + d];
        S[b * DD + d] = s;
    }
}

// --- Kernel 4: Y = LeakyReLU(S @ W + bias) via v_wmma_f32_16x16x4_f32 -------
// S: 16x64 (M=batch, K), W: 64x64 row-major (K x N), Y: 16x64.
// 1 block, 128 threads = 4 waves; wave w owns N-tile [16w, 16w+16).
// FP32 WMMA fragment layouts (ISA 7.12.2):
//   A 16x4:  lanes 0-15 -> {K0,K1} of row M=lane ; lanes 16-31 -> {K2,K3} of M=lane-16
//   B 4x16:  lanes 0-15 -> {K0,K1} of col N=lane ; lanes 16-31 -> {K2,K3} of N=lane-16
//   C 16x16: VGPR r: lanes 0-15 -> (M=r, N=lane); lanes 16-31 -> (M=r+8, N=lane-16)
__global__ void gemm_wmma_kernel(const float* __restrict__ S,
                                 const float* __restrict__ W,
                                 const float* __restrict__ bias,
                                 float* __restrict__ Y) {
    const int lane = threadIdx.x & 31;
    const int wv   = threadIdx.x >> 5;    // 0..3 -> N-tile
    const int nb   = wv * 16;
    const int row  = lane & 15;           // M (for A) / N (for B) within tile
    const int hi   = lane >> 4;           // 0: K{0,1}, 1: K{2,3}

    v8f c = {};
#pragma unroll
    for (int k0 = 0; k0 < DD; k0 += 4) {
        const int ka = k0 + 2 * hi;
        v2f a, b;
        a[0] = S[row * DD + ka];
        a[1] = S[row * DD + ka + 1];
        b[0] = W[ka * DD + nb + row];
        b[1] = W[(ka + 1) * DD + nb + row];
        c = __builtin_amdgcn_wmma_f32_16x16x4_f32(
                /*neg_a=*/false, a, /*neg_b=*/false, b,
                /*c_mod=*/(short)0, c, /*reuse_a=*/false, /*reuse_b=*/false);
    }

    const int col = nb + row;             // output column (d)
    const float bc = bias[col];
#pragma unroll
    for (int r = 0; r < 8; ++r) {
        const int m = r + 8 * hi;         // output row (batch)
        float y = c[r] + bc;
        y = (y >= 0.0f) ? y : 0.01f * y;
        Y[m * DD + col] = y;
    }
}

// --- Kernel 5: broadcast replicated outputs with float4 stores --------------
// out = [ new_node_set: 16*128*64 = 131072 f32 | new_adj: 16*128*128 = 262144 f32 ]
__global__ void broadcast_kernel(const float* __restrict__ Y,
                                 const float* __restrict__ adj_sum,
                                 float* __restrict__ out) {
    const int i4 = blockIdx.x * blockDim.x + threadIdx.x;   // float4 index
    if (i4 < 32768) {                       // node region: 131072 floats
        const int flat = i4 * 4;
        const int b = flat >> 13;           // /8192 (=128*64 per batch)
        const int d = flat & 63;            // aligned since flat % 4 == 0
        float4 v = *reinterpret_cast<const float4*>(Y + b * DD + d);
        *reinterpret_cast<float4*>(out + flat) = v;
    } else if (i4 < 98304) {                // adj region: 262144 floats
        const int flat = i4 * 4 - 131072;
        const int b = flat >> 14;           // /16384 (=128*128 per batch)
        const float s = adj_sum[b];
        float4 v = make_float4(s, s, s, s);
        *reinterpret_cast<float4*>(out + 131072 + flat) = v;
    }
}

extern "C" void kernel_launch(void* const* d_in, const int* in_sizes, int n_in,
                              void* d_out, int out_size, void* d_ws, size_t ws_size,
                              hipStream_t stream) {
    const float* node  = (const float*)d_in[0];
    const float* adj   = (const float*)d_in[1];
    // d_in[2] centroids, d_in[3] agg_w, d_in[4] agg_b: dead (softmax over
    // singleton axis => C == 1 exactly).
    const float* lin_w = (const float*)d_in[5];
    const float* lin_b = (const float*)d_in[6];

    float* ws       = (float*)d_ws;
    float* adj_sum  = ws + WS_ADJSUM;
    float* adj_part = ws + WS_ADJPART;
    float* S        = ws + WS_S;
    float* Y        = ws + WS_Y;
    float* out      = (float*)d_out;

    adj_partial_kernel<<<dim3(256, BB), 256, 0, stream>>>(adj, adj_part);
    adj_final_kernel  <<<BB, 256, 0, stream>>>(adj_part, adj_sum);
    node_rowsum_kernel<<<BB, 1024, 0, stream>>>(node, S);
    gemm_wmma_kernel  <<<1, 128, 0, stream>>>(S, lin_w, lin_b, Y);
    broadcast_kernel  <<<384, 256, 0, stream>>>(Y, adj_sum, out);
}